// VectorQuantizerEMA_89850715832448
// MI455X (gfx1250) — compile-verified
//
#include <hip/hip_runtime.h>
#include <hip/hip_bf16.h>

// Problem constants (match reference)
#define NTOK 16384            // 32*512 tokens
#define KCODE 8192            // num embeddings
#define DDIM 64               // embedding dim
#define DECAY_F 0.99f
#define EPS_F 1e-5f
#define BATCH_F 32.0f

typedef float v2f __attribute__((ext_vector_type(2)));
typedef float v8f __attribute__((ext_vector_type(8)));
typedef int   v4i __attribute__((ext_vector_type(4)));

#define AS1 __attribute__((address_space(1)))
#define AS3 __attribute__((address_space(3)))

// ---------------------------------------------------------------------------
// Kernel 1: per-code squared norm, zero the histogram, zero the x-sum
// accumulator (which lives in the new_weight output slot).
// ---------------------------------------------------------------------------
__global__ __launch_bounds__(256) void k_cnorm(const float* __restrict__ cb,
                                               float* __restrict__ cnorm,
                                               unsigned* __restrict__ hist,
                                               float* wsum_acc) {
    int k = blockIdx.x * blockDim.x + threadIdx.x;
    if (k >= KCODE) return;
    const float4* row = (const float4*)(cb + (size_t)k * DDIM);
    float4* wz = (float4*)(wsum_acc + (size_t)k * DDIM);
    float s = 0.0f;
    #pragma unroll
    for (int j = 0; j < DDIM / 4; ++j) {
        float4 v = row[j];
        s += v.x * v.x + v.y * v.y + v.z * v.z + v.w * v.w;
        wz[j] = make_float4(0.f, 0.f, 0.f, 0.f);
    }
    cnorm[k] = s;
    hist[k] = 0u;
}

// ---------------------------------------------------------------------------
// Kernel 2: fp32 WMMA distance GEMM + per-row argmin, with double-buffered
// async (ASYNCcnt-tracked) codebook tile staging into LDS.
// Block = 256 threads = 8 waves; wave owns 16 rows; block owns 128 rows.
// Tile j+1 streams into the alternate LDS buffer while tile j is consumed
// by 16 chained V_WMMA_F32_16X16X4_F32 per wave.
// Distance proxy: ||c||^2 - 2*x.c  (||x||^2 constant per row -> argmin same).
// ---------------------------------------------------------------------------
#define LDS_STRIDE 68            // pad 64 -> 68 floats: conflict-free ds_load_b64
#define NTILE (KCODE / 16)       // 512 codebook tiles

__device__ __forceinline__ void block_barrier() {
    __builtin_amdgcn_s_barrier_signal(-1);
    __builtin_amdgcn_s_barrier_wait(-1);
}

__global__ __launch_bounds__(256) void k_argmin(const float* __restrict__ x,
                                                const float* __restrict__ cb,
                                                const float* __restrict__ cnorm,
                                                int* __restrict__ idx_out) {
    __shared__ float ldsB[2][16 * LDS_STRIDE];

    const int t    = threadIdx.x;
    const int wave = t >> 5;        // 0..7
    const int lane = t & 31;        // wave32 lane
    const int col  = lane & 15;     // C/D layout: N = lane & 15
    const int hi   = lane >> 4;     // C/D layout: M = r + 8*hi
    const int rowBase = blockIdx.x * 128 + wave * 16;

    // --- Load A fragments: 16 rows x 64 cols of x, f32 16x4 fragment layout.
    // frag j, lane l: elements (M = l&15, kk = 4j + 2*(l>>4) + {0,1})
    v2f a[16];
    {
        const float* xrow = x + (size_t)(rowBase + col) * DDIM + 2 * hi;
        #pragma unroll
        for (int j = 0; j < 16; ++j) {
            a[j].x = xrow[4 * j];
            a[j].y = xrow[4 * j + 1];
        }
    }

    // --- Async tile staging setup: thread t owns a 16B chunk of each tile.
    const int code = t >> 4;        // 0..15 : code row within tile
    const int d4   = t & 15;        // 0..15 : float4 slot within 64-float row
    const float* gsrc = cb + (size_t)code * DDIM + (size_t)d4 * 4;
    AS3 v4i* ldst0 = (AS3 v4i*)&ldsB[0][code * LDS_STRIDE + d4 * 4];
    AS3 v4i* ldst1 = (AS3 v4i*)&ldsB[1][code * LDS_STRIDE + d4 * 4];

    // Prologue: tile 0 -> buffer 0
    __builtin_amdgcn_global_load_async_to_lds_b128((AS1 v4i*)gsrc, ldst0, 0, 0);

    float best[8];
    int   bidx[8];
    #pragma unroll
    for (int r = 0; r < 8; ++r) { best[r] = 3.4e38f; bidx[r] = 0; }

    float cn_cur = cnorm[col];      // ||c||^2 for tile 0, this lane's column

    for (int j = 0; j < NTILE; ++j) {
        const int c0 = j * 16;
        float cn_nxt = 0.0f;
        if (j + 1 < NTILE) {
            // Issue tile j+1 into the alternate buffer (overlaps this tile's
            // WMMA chain), prefetch its cnorm, then wait for the OLDER async
            // load only (in-order completion => tile j has landed).
            const float* gn = gsrc + (size_t)(j + 1) * (16 * DDIM);
            AS3 v4i* ln = (j & 1) ? ldst0 : ldst1;
            __builtin_amdgcn_global_load_async_to_lds_b128((AS1 v4i*)gn, ln, 0, 0);
            cn_nxt = cnorm[c0 + 16 + col];
            asm volatile("s_wait_asynccnt 0x1" ::: "memory");
        } else {
            asm volatile("s_wait_asynccnt 0x0" ::: "memory");
        }
        block_barrier();            // all waves' tile-j chunks are in LDS

        // 16x16 dot-product tile: D = sum_q A_q (16x4) * B_q (4x16)
        v8f acc = {0.f, 0.f, 0.f, 0.f, 0.f, 0.f, 0.f, 0.f};
        const float* brow = &ldsB[j & 1][col * LDS_STRIDE + 2 * hi];
        #pragma unroll
        for (int q = 0; q < 16; ++q) {
            v2f b;
            b.x = brow[4 * q];
            b.y = brow[4 * q + 1];
            acc = __builtin_amdgcn_wmma_f32_16x16x4_f32(
                false, a[q], false, b, (short)0, acc, false, false);
        }

        const int codeIdx = c0 + col;
        #pragma unroll
        for (int r = 0; r < 8; ++r) {
            float score = cn_cur - 2.0f * acc[r];
            if (score < best[r]) { best[r] = score; bidx[r] = codeIdx; }
        }
        cn_cur = cn_nxt;

        block_barrier();            // all waves done reading buf[j&1] before
                                    // the async issue of tile j+2 overwrites it
    }

    // Butterfly min-reduce across the 16-lane column group.
    // Tie-break toward lower code index to match jnp.argmin.
    #pragma unroll
    for (int m = 1; m < 16; m <<= 1) {
        #pragma unroll
        for (int r = 0; r < 8; ++r) {
            float ov = __shfl_xor(best[r], m, 32);
            int   oi = __shfl_xor(bidx[r], m, 32);
            if (ov < best[r] || (ov == best[r] && oi < bidx[r])) {
                best[r] = ov; bidx[r] = oi;
            }
        }
    }
    if (col == 0) {
        #pragma unroll
        for (int r = 0; r < 8; ++r)
            idx_out[rowBase + r + 8 * hi] = bidx[r];
    }
}

// ---------------------------------------------------------------------------
// Kernel 3: write one-hot rows of `discrete` (single coalesced pass),
// gather `quantized`, histogram counts, scatter x-sums into new_weight slot.
// Block = 256 threads = 4 rows x 64 lanes; grid = N/4.
// ---------------------------------------------------------------------------
__global__ __launch_bounds__(256) void k_scatter(const float* __restrict__ x,
                                                 const float* __restrict__ cb,
                                                 const int* __restrict__ idx,
                                                 float* __restrict__ discrete,
                                                 float* __restrict__ quant,
                                                 float* wsum_acc,
                                                 unsigned* __restrict__ hist) {
    const int t    = threadIdx.x;
    const int rsub = t >> 6;        // 0..3 row within block
    const int lane = t & 63;        // 0..63
    const int n    = blockIdx.x * 4 + rsub;
    const int id   = idx[n];

    // One-hot row: 8192 floats = 2048 float4, 64 lanes x 32 iterations,
    // lane-interleaved for fully coalesced 1KB stores.
    float4* drow = (float4*)(discrete + (size_t)n * KCODE);
    #pragma unroll
    for (int j = 0; j < 32; ++j) {
        const int f4 = j * 64 + lane;
        const int e0 = f4 * 4;
        float4 v = make_float4(0.f, 0.f, 0.f, 0.f);
        unsigned rel = (unsigned)(id - e0);
        if (rel < 4u) ((float*)&v)[rel] = 1.0f;
        drow[f4] = v;
    }

    quant[(size_t)n * DDIM + lane] = cb[(size_t)id * DDIM + lane];
    atomicAdd(&wsum_acc[(size_t)id * DDIM + lane], x[(size_t)n * DDIM + lane]);
    if (lane == 0) atomicAdd(&hist[id], 1u);
}

// ---------------------------------------------------------------------------
// Kernel 4: EMA finalize. One thread per (k,d). wsum accumulator aliases
// out_weight (each thread reads/writes only its own element).
// ---------------------------------------------------------------------------
__global__ __launch_bounds__(256) void k_finalize(const float* __restrict__ ema_count,
                                                  const float* __restrict__ ema_weight,
                                                  const unsigned* __restrict__ hist,
                                                  float* __restrict__ out_count,
                                                  float* out_weight,
                                                  float* __restrict__ out_codebook) {
    const int i = blockIdx.x * blockDim.x + threadIdx.x;   // 0 .. K*D-1
    const int k = i >> 6;

    const float h   = (float)hist[k];
    float cnt = ema_count[k] * DECAY_F + h * (1.0f - DECAY_F);
    cnt = (cnt + EPS_F) / (BATCH_F + (float)KCODE * EPS_F) * BATCH_F;

    const float nw = ema_weight[i] * DECAY_F + out_weight[i] * (1.0f - DECAY_F);
    out_weight[i]   = nw;
    out_codebook[i] = nw / cnt;
    if ((i & 63) == 0) out_count[k] = cnt;
}

// ---------------------------------------------------------------------------
extern "C" void kernel_launch(void* const* d_in, const int* in_sizes, int n_in,
                              void* d_out, int out_size, void* d_ws, size_t ws_size,
                              hipStream_t stream) {
    const float* x          = (const float*)d_in[0];   // [32,512,64]
    const float* codebook   = (const float*)d_in[1];   // [8192,64]
    const float* ema_count  = (const float*)d_in[2];   // [8192]
    const float* ema_weight = (const float*)d_in[3];   // [8192,64]

    float* out          = (float*)d_out;
    float* out_discrete = out;                                         // N*K
    float* out_quant    = out_discrete + (size_t)NTOK * KCODE;         // N*D
    float* out_count    = out_quant    + (size_t)NTOK * DDIM;          // K
    float* out_weight   = out_count    + KCODE;                        // K*D
    float* out_codebook = out_weight   + (size_t)KCODE * DDIM;         // K*D

    char* ws = (char*)d_ws;
    int*      idx   = (int*)ws;                                        // N ints
    unsigned* hist  = (unsigned*)(ws + (size_t)NTOK * sizeof(int));    // K u32
    float*    cnorm = (float*)(ws + (size_t)NTOK * sizeof(int)
                                  + (size_t)KCODE * sizeof(unsigned)); // K f32

    // 1) code norms + zero hist + zero x-sum accumulator (out_weight slot)
    k_cnorm<<<KCODE / 256, 256, 0, stream>>>(codebook, cnorm, hist, out_weight);
    // 2) WMMA distance GEMM + argmin (async double-buffered codebook tiles)
    k_argmin<<<NTOK / 128, 256, 0, stream>>>(x, codebook, cnorm, idx);
    // 3) one-hot + quantized gather + histogram + x-sum scatter
    k_scatter<<<NTOK / 4, 256, 0, stream>>>(x, codebook, idx, out_discrete,
                                            out_quant, out_weight, hist);
    // 4) EMA finalize
    k_finalize<<<(KCODE * DDIM) / 256, 256, 0, stream>>>(
        ema_count, ema_weight, hist, out_count, out_weight, out_codebook);
}